// HarModel_74792560492735
// MI455X (gfx1250) — compile-verified
//
#include <hip/hip_runtime.h>
#include <math.h>

// ---- problem constants (from reference) ----
#define TT      500
#define BB      8192
#define TAUc    0.005f
#define MAXINC  400.0f
#define MBc     1.2f
#define Gc      9.81f
#define EPSc    1e-12f
#define IRZZc   1.0e-4f
// scale(k, base) = (1 + (0.5 - logit)*MAX_CHANGE/100) * base ; MAX_CHANGE/100 = 0.95
#define SCL(v, base) ((1.0f + (0.5f - (v)) * 0.95f) * (base))

#define NB_L7   512   // blocks for logits[:,:,7] partial reduction
#define NB_SIM  (BB / 32)

typedef __attribute__((ext_vector_type(2))) float v2f;
typedef __attribute__((ext_vector_type(8))) float v8f;

// ---------------------------------------------------------------------------
// wave32 sum via V_WMMA_F32_16X16X4_F32:
//   A(16x4) holds the 32 lane partials (lane L<16 -> A[L,0], lane L>=16 -> A[L-16,2]),
//   B(4x16) = ones  =>  D[m,n] = acc_m + acc_{m+16}  (independent of n).
//   Lane L holds D rows (L<16 ? 0..7 : 8..15) in its 8 result VGPRs, so
//   sum(d[0..7]) + shfl_xor(.,16) = full 32-lane sum, all in f32.
// ---------------------------------------------------------------------------
__device__ __forceinline__ float wave32_sum_wmma(float x) {
    v2f a; a.x = x;    a.y = 0.0f;
    v2f b; b.x = 1.0f; b.y = 1.0f;
    v8f c = {};
    v8f d = __builtin_amdgcn_wmma_f32_16x16x4_f32(
        /*neg_a=*/false, a, /*neg_b=*/false, b,
        /*c_mod=*/(short)0, c, /*reuse_a=*/false, /*reuse_b=*/false);
    float s = d[0] + d[1] + d[2] + d[3] + d[4] + d[5] + d[6] + d[7];
    s += __shfl_xor(s, 16, 32);
    return s;
}

// ---------------------------------------------------------------------------
// Pass 1: deterministic partial sums of logits[:,:,7]  (also pre-warms L2
// with the logits tensor for the simulation pass).
// ---------------------------------------------------------------------------
__global__ __launch_bounds__(256) void k_l7_partial(const float* __restrict__ logits,
                                                    float* __restrict__ partials) {
    __shared__ float sm[256];
    const long long n = (long long)TT * BB;
    float s = 0.0f;
    for (long long i = (long long)blockIdx.x * blockDim.x + threadIdx.x; i < n;
         i += (long long)gridDim.x * blockDim.x) {
        s += logits[i * 12 + 7];
    }
    sm[threadIdx.x] = s;
    __syncthreads();
    for (int w = 128; w > 0; w >>= 1) {
        if ((int)threadIdx.x < w) sm[threadIdx.x] += sm[threadIdx.x + w];
        __syncthreads();
    }
    if (threadIdx.x == 0) partials[blockIdx.x] = sm[0];
}

// Pass 2: fixed-order combine -> hover (scalar needed by every sim lane)
__global__ void k_hover(const float* __restrict__ partials, float* __restrict__ hover_out) {
    float s = 0.0f;
    for (int i = 0; i < NB_L7; ++i) s += partials[i];
    const float mean_l7 = s / (float)((long long)TT * BB);
    const float kThm = SCL(mean_l7, 1.076e-5f);
    const float h2 = (MBc * Gc) / (4.0f * kThm + EPSc);
    hover_out[0] = sqrtf(fmaxf(h2, 1e-6f));
}

// ---------------------------------------------------------------------------
// Pass 3: per-lane 499-step integration; dead state (x, y, xd, yd) removed
// since it never feeds z. Software-pipelined loads + global prefetch.
// ---------------------------------------------------------------------------
__global__ __launch_bounds__(32) void k_sim(
    const float* __restrict__ labels, const float* __restrict__ logits,
    const float* __restrict__ u1p, const float* __restrict__ u2p,
    const float* __restrict__ u3p, const float* __restrict__ u4p,
    const float* __restrict__ mxp, const float* __restrict__ mnp,
    const float* __restrict__ hoverp, float* __restrict__ loss_part) {

    const int b = blockIdx.x * 32 + threadIdx.x;
    const float hover = hoverp[0];
    const float4* __restrict__ lp = (const float4*)logits;  // 3 float4 per (t,b)

    // live state
    float z = 0.0f, q0 = 1.0f, q1 = 0.0f, q2 = 0.0f, q3 = 0.0f;
    float zd = 0.0f, p = 0.0f, q = 0.0f, r = 0.0f;
    float w1 = hover, w2 = hover, w3 = hover, w4 = hover;
    float wd1 = 0.0f, wd2 = 0.0f, wd3 = 0.0f, wd4 = 0.0f;

    // loss term for t = 0: pred_z = 0
    float lab0 = labels[b];
    float lsum = lab0 * lab0;

    // prime the pipeline with t = 1 inputs
    int idx = BB + b;
    float4 L0 = lp[(size_t)idx * 3 + 0];
    float4 L1 = lp[(size_t)idx * 3 + 1];
    float4 L2 = lp[(size_t)idx * 3 + 2];
    float cu1 = u1p[idx], cu2 = u2p[idx], cu3 = u3p[idx], cu4 = u4p[idx];
    float cmn = mnp[idx], cmx = mxp[idx], clab = labels[idx];

    for (int t = 1; t < TT; ++t) {
        // -------- issue next-step loads first (hide latency behind compute)
        const int tn  = (t + 1 < TT) ? (t + 1) : t;
        const int idn = tn * BB + b;
        float4 N0 = lp[(size_t)idn * 3 + 0];
        float4 N1 = lp[(size_t)idn * 3 + 1];
        float4 N2 = lp[(size_t)idn * 3 + 2];
        float nu1 = u1p[idn], nu2 = u2p[idn], nu3 = u3p[idn], nu4 = u4p[idn];
        float nmn = mnp[idn], nmx = mxp[idn], nlab = labels[idn];

        // stream-ahead prefetch of the wide logits rows (global_prefetch_b8)
        int tpf = t + 6; if (tpf >= TT) tpf = TT - 1;
        __builtin_prefetch(logits + (size_t)(tpf * BB + b) * 12, 0, 0);

        // -------- unpack per-step parameters (logit index 2 is unused)
        const float dxm  = SCL(L0.x, 0.16f);
        const float dym  = SCL(L0.y, 0.16f);
        const float IBxx = SCL(L0.w, 0.0123f);
        const float IByy = SCL(L1.x, 0.0123f);
        const float IBzz = SCL(L1.y, 0.0123f);
        const float Cd   = SCL(L1.z, 0.1f);
        const float kTh  = SCL(L1.w, 1.076e-5f);
        const float kTo  = SCL(L2.x, 1.632e-7f);
        const float t2   = SCL(L2.y, 0.015f);
        const float kp_  = SCL(L2.z, 1.0f);
        const float dp   = SCL(L2.w, 1.0f);

        // -------- motor 2nd-order dynamics
        const float it2  = 1.0f / (t2 * t2);
        const float m2dt = -2.0f * dp * t2;
        const float wdd1 = m2dt * wd1 - w1 + kp_ * cu1 * it2;
        const float wdd2 = m2dt * wd2 - w2 + kp_ * cu2 * it2;
        const float wdd3 = m2dt * wd3 - w3 + kp_ * cu3 * it2;
        const float wdd4 = m2dt * wd4 - w4 + kp_ * cu4 * it2;

        const float w1c = fminf(fmaxf(w1, cmn), cmx);
        const float w2c = fminf(fmaxf(w2, cmn), cmx);
        const float w3c = fminf(fmaxf(w3, cmn), cmx);
        const float w4c = fminf(fmaxf(w4, cmn), cmx);

        const float Th1 = kTh * w1c * w1c, Th2 = kTh * w2c * w2c;
        const float Th3 = kTh * w3c * w3c, Th4 = kTh * w4c * w4c;
        const float To1 = kTo * w1c * w1c, To2 = kTo * w2c * w2c;
        const float To3 = kTo * w3c * w3c, To4 = kTo * w4c * w4c;
        const float ThrSum = Th1 + Th2 + Th3 + Th4;

        // -------- vertical velocity ( -Cd*sign(zd)*zd^2 == -Cd*zd*|zd| )
        const float nzd = zd + TAUc * (-Cd * zd * fabsf(zd)
                        + ThrSum * (q0 * q0 - q1 * q1 - q2 * q2 + q3 * q3)
                        - Gc * MBc) / MBc;

        // -------- quaternion kinematics
        const float nq0 = q0 + TAUc * (-0.5f * p * q1 - 0.5f * q * q2 - 0.5f * r * q3);
        const float nq1 = q1 + TAUc * ( 0.5f * p * q0 - 0.5f * q * q3 + 0.5f * r * q2);
        const float nq2 = q2 + TAUc * ( 0.5f * p * q3 + 0.5f * q * q0 - 0.5f * r * q1);
        const float nq3 = q3 + TAUc * (-0.5f * p * q2 + 0.5f * q * q1 + 0.5f * r * q0);

        // -------- body rates
        const float wsum = w1c - w2c + w3c - w4c;
        const float gp = -IRZZc * wsum * q;
        const float gq =  IRZZc * wsum * p;
        const float np_ = p + TAUc * ((IByy - IBzz) * q * r + gp + (Th1 - Th2 - Th3 + Th4) * dym) / IBxx;
        const float nq_ = q + TAUc * ((IBzz - IBxx) * p * r + gq + (Th1 + Th2 - Th3 - Th4) * dxm) / IByy;
        const float nr_ = r + TAUc * ((IBxx - IByy) * p * q - To1 + To2 - To3 + To4) / IBzz;

        // -------- position (z only), motors
        const float cand = z + TAUc * nzd;
        const float nw1 = w1c + TAUc * wd1, nw2 = w2c + TAUc * wd2;
        const float nw3 = w3c + TAUc * wd3, nw4 = w4c + TAUc * wd4;
        const float nwd1 = wd1 + TAUc * wdd1, nwd2 = wd2 + TAUc * wdd2;
        const float nwd3 = wd3 + TAUc * wdd3, nwd4 = wd4 + TAUc * wdd4;

        // mask: isfinite(cand) && |cand - z| <= MAX_INC  (NaN/Inf both fail the compare)
        const float nz = (fabsf(cand - z) <= MAXINC) ? cand : z;

        // -------- loss accumulation
        const float diff = nz - clab;
        lsum = fmaf(diff, diff, lsum);

        // -------- commit state, rotate pipelined inputs
        z = nz; q0 = nq0; q1 = nq1; q2 = nq2; q3 = nq3;
        zd = nzd; p = np_; q = nq_; r = nr_;
        w1 = nw1; w2 = nw2; w3 = nw3; w4 = nw4;
        wd1 = nwd1; wd2 = nwd2; wd3 = nwd3; wd4 = nwd4;
        L0 = N0; L1 = N1; L2 = N2;
        cu1 = nu1; cu2 = nu2; cu3 = nu3; cu4 = nu4;
        cmn = nmn; cmx = nmx; clab = nlab;
    }

    // wave32 reduction via WMMA (uniform control flow, EXEC all ones)
    const float wsum_loss = wave32_sum_wmma(lsum);
    if (threadIdx.x == 0) loss_part[blockIdx.x] = wsum_loss;
}

// Pass 4: fixed-order combine of per-wave partials -> scalar MSE
__global__ void k_final(const float* __restrict__ part, float* __restrict__ out) {
    float s = 0.0f;
    for (int i = 0; i < NB_SIM; ++i) s += part[i];
    out[0] = s / (float)((long long)TT * BB);
}

// ---------------------------------------------------------------------------
extern "C" void kernel_launch(void* const* d_in, const int* in_sizes, int n_in,
                              void* d_out, int out_size, void* d_ws, size_t ws_size,
                              hipStream_t stream) {
    const float* labels = (const float*)d_in[0];
    const float* logits = (const float*)d_in[1];
    const float* u1     = (const float*)d_in[2];
    const float* u2     = (const float*)d_in[3];
    const float* u3     = (const float*)d_in[4];
    const float* u4     = (const float*)d_in[5];
    const float* mx     = (const float*)d_in[6];   // maxMotor
    const float* mn     = (const float*)d_in[7];   // minMotor
    float* out = (float*)d_out;

    float* ws        = (float*)d_ws;
    float* hover     = ws;            // 1 float
    float* l7part    = ws + 64;       // NB_L7 floats (aligned region)
    float* losspart  = ws + 64 + NB_L7; // NB_SIM floats

    hipLaunchKernelGGL(k_l7_partial, dim3(NB_L7), dim3(256), 0, stream, logits, l7part);
    hipLaunchKernelGGL(k_hover, dim3(1), dim3(1), 0, stream, l7part, hover);
    hipLaunchKernelGGL(k_sim, dim3(NB_SIM), dim3(32), 0, stream,
                       labels, logits, u1, u2, u3, u4, mx, mn, hover, losspart);
    hipLaunchKernelGGL(k_final, dim3(1), dim3(1), 0, stream, losspart, out);
}